// GCNConv_3023656976832
// MI455X (gfx1250) — compile-verified
//
#include <hip/hip_runtime.h>

// GCN layer on MI455X (gfx1250), single fused kernel:
//   out = (D^-1 * A * X) @ W        (== D^-1 * A * (X @ W) by linearity)
// Phase 1: per-wave neighbor gather+average of X rows into LDS (L2-bound part).
// Phase 2: 16x128 @ 128x128 fp32 GEMM per block via V_WMMA_F32_16X16X4_F32.

#define DEG   64
#define DDIM  128
#define LDSP  132   // padded LDS row stride (floats) -> conflict-free WMMA A loads

typedef __attribute__((ext_vector_type(2))) float v2f;
typedef __attribute__((ext_vector_type(8))) float v8f;

__global__ __launch_bounds__(256) void gcn_fused_kernel(
    const float* __restrict__ X,      // [n, 128]
    const float* __restrict__ W,      // [128, 128]
    const int*   __restrict__ rowptr, // [n+1] (regular CSR; unused, deg fixed)
    const int*   __restrict__ cols,   // [n*64]
    const float* __restrict__ degs,   // [n]
    float*       __restrict__ out,    // [n, 128]
    int n)
{
    __shared__ float sX[16 * LDSP];   // aggregated 16x128 tile (padded)

    const int tid  = threadIdx.x;
    const int wave = tid >> 5;        // 0..7
    const int lane = tid & 31;
    const int m0   = blockIdx.x * 16;
    (void)rowptr;

    // ---------------- Phase 1: gather + average neighbors (2 nodes / wave) ----
    #pragma unroll
    for (int t = 0; t < 2; ++t) {
        const int local = wave * 2 + t;     // 0..15
        const int g     = m0 + local;       // global node id
        float4 acc = make_float4(0.f, 0.f, 0.f, 0.f);
        if (g < n) {
            // 64 neighbor indices: 2 per lane, broadcast via shuffle
            const int2 myidx = *(const int2*)(cols + g * DEG + lane * 2);
            #pragma unroll 8
            for (int e = 0; e < DEG; ++e) {
                const int held = (e & 1) ? myidx.y : myidx.x;
                const int col  = __shfl(held, e >> 1, 32);
                const float4 r = *(const float4*)(X + col * DDIM + lane * 4);
                acc.x += r.x; acc.y += r.y; acc.z += r.z; acc.w += r.w;
            }
            const float s = 1.0f / degs[g];
            acc.x *= s; acc.y *= s; acc.z *= s; acc.w *= s;
        }
        *(float4*)(sX + local * LDSP + lane * 4) = acc;
    }
    __syncthreads();

    // ---------------- Phase 2: out[m0:m0+16, :] = sX(16x128) @ W(128x128) ----
    // Wave w computes columns [16w, 16w+16). K-loop: 32 x v_wmma_f32_16x16x4_f32.
    const int half = lane >> 4;          // 0|1
    const int l16  = lane & 15;
    const int ncol = wave * 16 + l16;    // output column for this lane

    v8f c = {};
    #pragma unroll
    for (int kb = 0; kb < DDIM; kb += 4) {
        const int k0 = kb + half * 2;
        // A (16x4): a[v] = Xagg[l16][k0+v]  -> contiguous float2 from LDS
        const v2f a = *(const v2f*)(sX + l16 * LDSP + k0);
        // B (4x16): b[v] = W[k0+v][ncol]
        v2f b;
        b.x = W[k0 * DDIM + ncol];
        b.y = W[(k0 + 1) * DDIM + ncol];
        c = __builtin_amdgcn_wmma_f32_16x16x4_f32(
                /*neg_a=*/false, a, /*neg_b=*/false, b,
                /*c_mod=*/(short)0, c, /*reuse_a=*/false, /*reuse_b=*/false);
    }

    // C/D layout: VGPR v holds row (v + 8*half), col = l16
    #pragma unroll
    for (int v = 0; v < 8; ++v) {
        const int row = m0 + half * 8 + v;
        if (row < n) out[row * DDIM + ncol] = c[v];
    }
}

extern "C" void kernel_launch(void* const* d_in, const int* in_sizes, int n_in,
                              void* d_out, int out_size, void* d_ws, size_t ws_size,
                              hipStream_t stream) {
    (void)n_in; (void)out_size; (void)d_ws; (void)ws_size;
    const float* X      = (const float*)d_in[0];
    const float* W      = (const float*)d_in[1];
    const int*   rowptr = (const int*)d_in[2];
    const int*   cols   = (const int*)d_in[3];
    const float* degs   = (const float*)d_in[4];
    float*       out    = (float*)d_out;

    const int n      = in_sizes[4];          // N_NODES (degrees count)
    const int blocks = (n + 15) / 16;        // 625 for N=10000

    hipLaunchKernelGGL(gcn_fused_kernel, dim3(blocks), dim3(256), 0, stream,
                       X, W, rowptr, cols, degs, out, n);
}